// Postprocessing_torch_53961969107562
// MI455X (gfx1250) — compile-verified
//
#include <hip/hip_runtime.h>
#include <cstdint>

#define C_NUM   80
#define HH      512
#define WW      512
#define TILE_H  16
#define CHUNKS  (HH / TILE_H)      // 32
#define NBLK    (C_NUM * CHUNKS)   // 2560
#define KTOP    10
#define THREADS 256

#define AS1 __attribute__((address_space(1)))
#define AS3 __attribute__((address_space(3)))

typedef __attribute__((ext_vector_type(4))) int v4i;

#ifndef __has_builtin
#define __has_builtin(x) 0
#endif

#if __has_builtin(__builtin_amdgcn_global_load_async_to_lds_b128)
#define HAVE_ASYNC_LDS 1
#else
#define HAVE_ASYNC_LDS 0
#endif

// 16-byte async copy: HBM -> LDS, bypassing VGPRs (CDNA5 ASYNCcnt path).
__device__ __forceinline__ void async_copy_b128(const float* g, float* l) {
#if HAVE_ASYNC_LDS
  __builtin_amdgcn_global_load_async_to_lds_b128(
      (AS1 v4i*)g, (AS3 v4i*)l, /*offset=*/0, /*cpol=*/0);
#else
  // Fallback (synchronous) if the builtin is unavailable on this toolchain.
  float4 v = *(const float4*)g;
  *(float4*)l = v;
#endif
}

__device__ __forceinline__ void wait_async() {
#if HAVE_ASYNC_LDS
#if __has_builtin(__builtin_amdgcn_s_wait_asynccnt)
  __builtin_amdgcn_s_wait_asynccnt(0);
#else
  asm volatile("s_wait_asynccnt 0" ::: "memory");
#endif
#endif
}

// Lexicographic "better": higher score wins; ties -> smaller flat index
// (matches lax.top_k tie-breaking).
__device__ __forceinline__ bool better(float s1, int i1, float s2, int i2) {
  return (s1 > s2) || ((s1 == s2) && (i1 < i2));
}

// ---------------------------------------------------------------------------
// Kernel 1: per (class, 16-row chunk): 3x3 NMS from an async-staged LDS tile,
// then block-local top-10 -> workspace.
// ---------------------------------------------------------------------------
__global__ __launch_bounds__(THREADS)
void nms_topk_partial(const float* __restrict__ kp,
                      float* __restrict__ outScore,
                      int* __restrict__ outIdx) {
  __shared__ float smem[(TILE_H + 2) * WW];  // 36864 B tile, later reused for merge
  const int tid   = threadIdx.x;
  const int cls   = blockIdx.x / CHUNKS;     // adjacent blocks share halo rows (L2)
  const int chunk = blockIdx.x % CHUNKS;
  const int row0  = chunk * TILE_H;

  // Stage rows [row0-1, row0+TILE_H] of this class plane into LDS (async).
  // 18 rows * 128 16B-chunks = 2304 copies; 9 per thread.
  for (int t = tid; t < (TILE_H + 2) * (WW / 4); t += THREADS) {
    const int ry = t >> 7;             // 0..17
    const int xc = (t & 127) << 2;     // float column, multiple of 4
    const int gy = row0 - 1 + ry;
    float* l = &smem[ry * WW + xc];
    if (gy >= 0 && gy < HH) {
      const float* g = kp + ((size_t)cls * HH + (size_t)gy) * WW + xc;
      async_copy_b128(g, l);
    } else {
      // SAME padding: values are >= 0, so 0-fill is equivalent to -inf pad here.
      l[0] = 0.f; l[1] = 0.f; l[2] = 0.f; l[3] = 0.f;
    }
  }
  wait_async();
  __syncthreads();

  // Per-thread top-10 over this thread's 32 elements.
  float ts[KTOP];
  int   ti[KTOP];
#pragma unroll
  for (int j = 0; j < KTOP; ++j) { ts[j] = -1.0f; ti[j] = 0x7fffffff; }

  for (int e = tid; e < TILE_H * WW; e += THREADS) {
    const int x  = e & (WW - 1);
    const int yl = e >> 9;             // log2(WW)
    const float* r0 = &smem[yl * WW];
    const float* r1 = r0 + WW;
    const float* r2 = r1 + WW;
    const float v = r1[x];
    float m = fmaxf(v, fmaxf(r0[x], r2[x]));
    if (x > 0)      m = fmaxf(m, fmaxf(r1[x - 1], fmaxf(r0[x - 1], r2[x - 1])));
    if (x < WW - 1) m = fmaxf(m, fmaxf(r1[x + 1], fmaxf(r0[x + 1], r2[x + 1])));
    const float s  = (v == m) ? v : 0.0f;   // keypoint * keep
    const int  idx = ((row0 + yl) * WW + x) * C_NUM + cls;
    if (better(s, idx, ts[KTOP - 1], ti[KTOP - 1])) {
      int j = KTOP - 1;
      while (j > 0 && better(s, idx, ts[j - 1], ti[j - 1])) {
        ts[j] = ts[j - 1]; ti[j] = ti[j - 1]; --j;
      }
      ts[j] = s; ti[j] = idx;
    }
  }
  __syncthreads();  // done reading tile; reuse LDS for the merge

  float* ms = smem;                                  // 2560 scores
  int*   mi = (int*)(smem + THREADS * KTOP);         // 2560 indices
  float* rs = smem + 2 * THREADS * KTOP;             // 256
  int*   ri = (int*)(smem + 2 * THREADS * KTOP + THREADS);
  int*   rp = (int*)(smem + 2 * THREADS * KTOP + 2 * THREADS);

#pragma unroll
  for (int j = 0; j < KTOP; ++j) {
    ms[tid * KTOP + j] = ts[j];
    mi[tid * KTOP + j] = ti[j];
  }
  __syncthreads();

  for (int k = 0; k < KTOP; ++k) {
    float bs = -2.0f; int bi = 0x7fffffff; int bp = 0;
    for (int t = tid; t < THREADS * KTOP; t += THREADS)
      if (better(ms[t], mi[t], bs, bi)) { bs = ms[t]; bi = mi[t]; bp = t; }
    rs[tid] = bs; ri[tid] = bi; rp[tid] = bp;
    __syncthreads();
    for (int off = THREADS >> 1; off > 0; off >>= 1) {
      if (tid < off && better(rs[tid + off], ri[tid + off], rs[tid], ri[tid])) {
        rs[tid] = rs[tid + off]; ri[tid] = ri[tid + off]; rp[tid] = rp[tid + off];
      }
      __syncthreads();
    }
    if (tid == 0) {
      outScore[blockIdx.x * KTOP + k] = rs[0];
      outIdx[blockIdx.x * KTOP + k]   = ri[0];
      ms[rp[0]] = -2.0f; mi[rp[0]] = 0x7fffffff;   // consume winner
    }
    __syncthreads();
  }
}

// ---------------------------------------------------------------------------
// Kernel 2: reduce 2560*10 candidates to global top-10, then box math.
// ---------------------------------------------------------------------------
__global__ __launch_bounds__(THREADS)
void final_topk_boxes(const float* __restrict__ wss, const int* __restrict__ wsi,
                      const float* __restrict__ offp, const float* __restrict__ szp,
                      float* __restrict__ out) {
  __shared__ float ms[THREADS * KTOP];
  __shared__ int   mi[THREADS * KTOP];
  __shared__ float rs[THREADS];
  __shared__ int   ri[THREADS];
  __shared__ int   rp[THREADS];
  __shared__ float fs[KTOP];
  __shared__ int   fi[KTOP];

  const int tid = threadIdx.x;
  float ts[KTOP];
  int   ti[KTOP];
#pragma unroll
  for (int j = 0; j < KTOP; ++j) { ts[j] = -1.0f; ti[j] = 0x7fffffff; }

  const int N = NBLK * KTOP;  // 25600
  for (int t = tid; t < N; t += THREADS) {
    const float s  = wss[t];
    const int  idx = wsi[t];
    if (better(s, idx, ts[KTOP - 1], ti[KTOP - 1])) {
      int j = KTOP - 1;
      while (j > 0 && better(s, idx, ts[j - 1], ti[j - 1])) {
        ts[j] = ts[j - 1]; ti[j] = ti[j - 1]; --j;
      }
      ts[j] = s; ti[j] = idx;
    }
  }
#pragma unroll
  for (int j = 0; j < KTOP; ++j) {
    ms[tid * KTOP + j] = ts[j];
    mi[tid * KTOP + j] = ti[j];
  }
  __syncthreads();

  for (int k = 0; k < KTOP; ++k) {
    float bs = -2.0f; int bi = 0x7fffffff; int bp = 0;
    for (int t = tid; t < THREADS * KTOP; t += THREADS)
      if (better(ms[t], mi[t], bs, bi)) { bs = ms[t]; bi = mi[t]; bp = t; }
    rs[tid] = bs; ri[tid] = bi; rp[tid] = bp;
    __syncthreads();
    for (int off = THREADS >> 1; off > 0; off >>= 1) {
      if (tid < off && better(rs[tid + off], ri[tid + off], rs[tid], ri[tid])) {
        rs[tid] = rs[tid + off]; ri[tid] = ri[tid + off]; rp[tid] = rp[tid + off];
      }
      __syncthreads();
    }
    if (tid == 0) {
      fs[k] = rs[0]; fi[k] = ri[0];
      ms[rp[0]] = -2.0f; mi[rp[0]] = 0x7fffffff;
    }
    __syncthreads();
  }

  if (tid < KTOP) {
    const int   k     = tid;
    const int   idx   = fi[k];
    const float score = fs[k];
    const int   chan  = idx / C_NUM;
    const int   cls   = idx - chan * C_NUM;
    const float y_idx = floorf((float)chan / 512.0f);
    const float x_idx = (float)chan - y_idx * 512.0f;
    const int   yi = (int)y_idx;
    const int   xi = (int)x_idx;
    const int   p  = yi * WW + xi;
    const float off0 = offp[p];             // channel 0
    const float off1 = offp[HH * WW + p];   // channel 1
    const float sz0  = szp[p];
    const float sz1  = szp[HH * WW + p];
    // reference flips the (ch0,ch1) pair -> (ch1,ch0)
    const float py = y_idx + off1;
    const float px = x_idx + off0;
    const float hy = sz1 * 0.5f;
    const float hx = sz0 * 0.5f;
    const float b0 = fminf(fmaxf(py - hy, 0.0f), 511.0f) * 4.0f;
    const float b1 = fminf(fmaxf(px - hx, 0.0f), 511.0f) * 4.0f;
    const float b2 = fminf(fmaxf(py + hy, 0.0f), 511.0f) * 4.0f;
    const float b3 = fminf(fmaxf(px + hx, 0.0f), 511.0f) * 4.0f;
    out[k * 4 + 0] = b0;
    out[k * 4 + 1] = b1;
    out[k * 4 + 2] = b2;
    out[k * 4 + 3] = b3;
    out[4 * KTOP + k]     = (float)cls;   // detection_classes
    out[5 * KTOP + k]     = score;        // detection_scores
  }
}

extern "C" void kernel_launch(void* const* d_in, const int* in_sizes, int n_in,
                              void* d_out, int out_size, void* d_ws, size_t ws_size,
                              hipStream_t stream) {
  const float* offset = (const float*)d_in[0];  // [1,2,512,512]
  const float* size_  = (const float*)d_in[1];  // [1,2,512,512]
  const float* kp     = (const float*)d_in[2];  // [1,80,512,512]

  float* wss = (float*)d_ws;                                       // 25600 scores
  int*   wsi = (int*)((char*)d_ws + sizeof(float) * NBLK * KTOP);  // 25600 indices

  nms_topk_partial<<<NBLK, THREADS, 0, stream>>>(kp, wss, wsi);
  final_topk_boxes<<<1, THREADS, 0, stream>>>(wss, wsi, offset, size_, (float*)d_out);
}